// DiseaseGNN_25366076850454
// MI455X (gfx1250) — compile-verified
//
#include <hip/hip_runtime.h>

// ---------------------------------------------------------------------------
// CDNA5 (gfx1250) wave32 WMMA implementation of the DiseaseGNN pipeline.
//
// Algebraic restructure vs. the naive graph:
//  * Layer 2: transform-before-scatter. mean(h1[src])@W^T == scatter(h1@W^T)/cnt,
//    so we scatter the 64-wide transformed rows instead of 128-wide h1 rows:
//    E*64 atomics instead of E*128.
//  * Edge head: the first MLP layer distributes over concat(h2[s],h2[d]):
//    hidden = relu(u[s] + v[d]),  u = h2@Wa^T + bc1,  v = h2@Wb^T.
//    Two N-scale GEMMs (1.6 GFLOP) replace an E-scale GEMM (26 GFLOP); the
//    per-edge work collapses to two coalesced 256B row reads + a 64->2 GEMV.
// All GEMMs use V_WMMA_F32_16X16X4_F32 (full fp32 matrix pipe).
// ---------------------------------------------------------------------------

typedef __attribute__((ext_vector_type(2))) float v2f;
typedef __attribute__((ext_vector_type(8))) float v8f;

__device__ __forceinline__ v8f wmma_f32_16x16x4(v2f a, v2f b, v8f c) {
  // 8 args: (neg_a, A, neg_b, B, c_mod, C, reuse_a, reuse_b)
  return __builtin_amdgcn_wmma_f32_16x16x4_f32(false, a, false, b, (short)0, c,
                                               false, false);
}

__device__ __forceinline__ void atomic_add_f32(float* p, float v) {
  __hip_atomic_fetch_add(p, v, __ATOMIC_RELAXED, __HIP_MEMORY_SCOPE_AGENT);
}

// ---------------------------------------------------------------------------
__global__ void zero4_kernel(float4* __restrict__ p, int n4) {
  int i = blockIdx.x * blockDim.x + threadIdx.x;
  if (i < n4) p[i] = make_float4(0.f, 0.f, 0.f, 0.f);
}

__global__ void degree_kernel(const long long* __restrict__ dst,
                              float* __restrict__ cnt, int E) {
  int e = blockIdx.x * blockDim.x + threadIdx.x;
  if (e < E) atomic_add_f32(&cnt[(int)dst[e]], 1.0f);
}

__global__ void invcnt_kernel(float* __restrict__ cnt, int N) {
  int i = blockIdx.x * blockDim.x + threadIdx.x;
  if (i < N) cnt[i] = 1.0f / fmaxf(cnt[i], 1.0f);
}

// Gather feat[src] rows (float4 chunks) and atomic-scatter-add into agg[dst].
// Only 64-wide scatters remain after the layer-2 restructure.
template <int D>
__global__ void scatter_kernel(const float* __restrict__ feat,
                               const long long* __restrict__ src,
                               const long long* __restrict__ dst,
                               float* __restrict__ agg, int E) {
  const int CH = D / 4;
  int t = blockIdx.x * blockDim.x + threadIdx.x;
  if (t >= E * CH) return;
  int e = t / CH, c = t % CH;
  long s = (long)src[e], d = (long)dst[e];
  float4 v = *(const float4*)(feat + s * (long)D + c * 4);
  float* o = agg + d * (long)D + c * 4;
  atomic_add_f32(o + 0, v.x);
  atomic_add_f32(o + 1, v.y);
  atomic_add_f32(o + 2, v.z);
  atomic_add_f32(o + 3, v.w);
}

// ---------------------------------------------------------------------------
// WMMA fragment convention (ISA 7.12.2, fp32 16x16x4):
//   A: lane&15 = row M, lane>>4 selects K pair {0,1}/{2,3} in the 2 VGPRs.
//   B mirrors A with N = lane&15.
//   C/D: VGPR r holds row (r + 8*(lane>>4)), col = lane&15.
// One wave computes a 16-node x 16-output tile.

// Layer 1: h_out = relu( (agg*inv) @ Wl^T + bias + x @ Wr^T ), fused dual GEMM.
template <int K, int OUTD>
__global__ void sage_gemm_kernel(const float* __restrict__ agg,
                                 const float* __restrict__ invc,
                                 const float* __restrict__ xin,
                                 const float* __restrict__ wl,
                                 const float* __restrict__ wr,
                                 const float* __restrict__ bias,
                                 float* __restrict__ hout, int nNodes) {
  const int lane = threadIdx.x & 31;
  const int wave = threadIdx.x >> 5;
  const int OT = OUTD / 16;
  const int w = blockIdx.x * (blockDim.x >> 5) + wave;
  const int nodeTile = w / OT, outTile = w % OT;
  const int nodeBase = nodeTile * 16;
  if (nodeBase >= nNodes) return;

  const int rc = lane & 15;
  const int kh = lane >> 4;
  const long arow = (long)(nodeBase + rc) * K + kh * 2;
  const long brow = (long)(outTile * 16 + rc) * K + kh * 2;
  const float inv = invc[nodeBase + rc];

  v8f acc = {};
#pragma unroll 8
  for (int k = 0; k < K; k += 4) {
    v2f a = *(const v2f*)(agg + arow + k);
    a = a * inv;  // fold mean scaling into A fragment (row-constant per lane)
    v2f b = *(const v2f*)(wl + brow + k);
    acc = wmma_f32_16x16x4(a, b, acc);
  }
#pragma unroll 8
  for (int k = 0; k < K; k += 4) {
    v2f a = *(const v2f*)(xin + arow + k);
    v2f b = *(const v2f*)(wr + brow + k);
    acc = wmma_f32_16x16x4(a, b, acc);
  }

  const int col = outTile * 16 + rc;
  const float bv = bias[col];
#pragma unroll
  for (int r = 0; r < 8; ++r) {
    int m = nodeBase + r + kh * 8;
    hout[(long)m * OUTD + col] = fmaxf(acc[r] + bv, 0.0f);
  }
}

// Plain GEMM: out = xin @ W^T (+ bias). W rows have runtime stride `bstride`
// so we can slice wc1 into Wa (cols 0..63) and Wb (cols 64..127).
template <int K, int OUTD, bool BIAS>
__global__ void gemm_plain_kernel(const float* __restrict__ xin,
                                  const float* __restrict__ w, int bstride,
                                  const float* __restrict__ bias,
                                  float* __restrict__ out, int nNodes) {
  const int lane = threadIdx.x & 31;
  const int wave = threadIdx.x >> 5;
  const int OT = OUTD / 16;
  const int gw = blockIdx.x * (blockDim.x >> 5) + wave;
  const int nodeTile = gw / OT, outTile = gw % OT;
  const int nodeBase = nodeTile * 16;
  if (nodeBase >= nNodes) return;

  const int rc = lane & 15;
  const int kh = lane >> 4;
  const long arow = (long)(nodeBase + rc) * K + kh * 2;
  const long brow = (long)(outTile * 16 + rc) * bstride + kh * 2;

  v8f acc = {};
#pragma unroll 8
  for (int k = 0; k < K; k += 4) {
    v2f a = *(const v2f*)(xin + arow + k);
    v2f b = *(const v2f*)(w + brow + k);
    acc = wmma_f32_16x16x4(a, b, acc);
  }

  const int col = outTile * 16 + rc;
  const float bv = BIAS ? bias[col] : 0.0f;
#pragma unroll
  for (int r = 0; r < 8; ++r) {
    int m = nodeBase + r + kh * 8;
    out[(long)m * OUTD + col] = acc[r] + bv;
  }
}

// Layer 2 (post-scatter): h2 = relu( h1 @ w2r^T + agg2*inv + bias ).
// K = 128, OUTD = 64. agg2 already holds scatter(h1 @ w2l^T).
__global__ void sage_gemm2_kernel(const float* __restrict__ h1,
                                  const float* __restrict__ w2r,
                                  const float* __restrict__ agg2,
                                  const float* __restrict__ invc,
                                  const float* __restrict__ bias,
                                  float* __restrict__ h2, int nNodes) {
  constexpr int K = 128, OUTD = 64;
  const int lane = threadIdx.x & 31;
  const int wave = threadIdx.x >> 5;
  const int OT = OUTD / 16;
  const int gw = blockIdx.x * (blockDim.x >> 5) + wave;
  const int nodeTile = gw / OT, outTile = gw % OT;
  const int nodeBase = nodeTile * 16;
  if (nodeBase >= nNodes) return;

  const int rc = lane & 15;
  const int kh = lane >> 4;
  const long arow = (long)(nodeBase + rc) * K + kh * 2;
  const long brow = (long)(outTile * 16 + rc) * K + kh * 2;

  v8f acc = {};
#pragma unroll 8
  for (int k = 0; k < K; k += 4) {
    v2f a = *(const v2f*)(h1 + arow + k);
    v2f b = *(const v2f*)(w2r + brow + k);
    acc = wmma_f32_16x16x4(a, b, acc);
  }

  const int col = outTile * 16 + rc;
  const float bv = bias[col];
#pragma unroll
  for (int r = 0; r < 8; ++r) {
    int m = nodeBase + r + kh * 8;
    float mean = agg2[(long)m * OUTD + col] * invc[m];
    h2[(long)m * OUTD + col] = fmaxf(acc[r] + mean + bv, 0.0f);
  }
}

// ---------------------------------------------------------------------------
// Edge head: one wave per edge.
//   hidden = relu(u[src] + v[dst])          (bc1 pre-folded into u)
//   out    = hidden @ wc2^T + bc2           (64->2 GEMV, wave32 butterfly)
// Each wave reads two contiguous 256B rows -> fully coalesced L2-resident hits.
__global__ void edge_head_kernel(const float* __restrict__ u,
                                 const float* __restrict__ v,
                                 const long long* __restrict__ src,
                                 const long long* __restrict__ dst,
                                 const float* __restrict__ wc2,
                                 const float* __restrict__ bc2,
                                 float* __restrict__ out, int E) {
  const int lane = threadIdx.x & 31;
  const int wave = threadIdx.x >> 5;
  const int e = blockIdx.x * (blockDim.x >> 5) + wave;
  if (e >= E) return;

  const long s = (long)src[e] * 64;
  const long d = (long)dst[e] * 64;
  v2f uu = *(const v2f*)(u + s + lane * 2);
  v2f vv = *(const v2f*)(v + d + lane * 2);
  float h0 = fmaxf(uu.x + vv.x, 0.0f);
  float h1 = fmaxf(uu.y + vv.y, 0.0f);
  v2f w0 = *(const v2f*)(wc2 + lane * 2);
  v2f w1 = *(const v2f*)(wc2 + 64 + lane * 2);
  float p0 = h0 * w0.x + h1 * w0.y;
  float p1 = h0 * w1.x + h1 * w1.y;
#pragma unroll
  for (int off = 16; off; off >>= 1) {
    p0 += __shfl_xor(p0, off, 32);
    p1 += __shfl_xor(p1, off, 32);
  }
  if (lane == 0) {
    float2 r = make_float2(p0 + bc2[0], p1 + bc2[1]);
    *(float2*)(out + (long)e * 2) = r;
  }
}

// ---------------------------------------------------------------------------
extern "C" void kernel_launch(void* const* d_in, const int* in_sizes, int n_in,
                              void* d_out, int out_size, void* d_ws,
                              size_t ws_size, hipStream_t stream) {
  (void)n_in; (void)out_size; (void)ws_size;
  const float* x = (const float*)d_in[0];
  const long long* ei = (const long long*)d_in[1];  // int64 edge_index
  const float* w1l = (const float*)d_in[2];
  const float* b1l = (const float*)d_in[3];
  const float* w1r = (const float*)d_in[4];
  const float* w2l = (const float*)d_in[5];
  const float* b2l = (const float*)d_in[6];
  const float* w2r = (const float*)d_in[7];
  const float* wc1 = (const float*)d_in[8];
  const float* bc1 = (const float*)d_in[9];
  const float* wc2 = (const float*)d_in[10];
  const float* bc2 = (const float*)d_in[11];

  const int N = in_sizes[0] / 64;  // 100000
  const int E = in_sizes[1] / 2;   // 1600000
  const long long* src = ei;
  const long long* dst = ei + E;

  // Workspace (floats): cnt[N] | bufA[N*64] | h1[N*128] | agg2[N*64] | h2[N*64]
  //   bufA: agg1, later reused for yl = h1 @ w2l^T
  //   h1 region reused for (u | v) after layer 2 consumes h1
  float* ws = (float*)d_ws;
  float* cnt = ws;
  float* bufA = ws + N;
  float* h1 = bufA + (long)N * 64;
  float* agg2 = h1 + (long)N * 128;
  float* h2 = agg2 + (long)N * 64;
  float* u = h1;
  float* v = h1 + (long)N * 64;
  float* out = (float*)d_out;

  const int B = 256;

  // ---- Layer 1: agg1 = scatter(x); h1 = relu(mean@w1l^T + b1l + x@w1r^T) ----
  zero4_kernel<<<(N / 4 + B - 1) / B, B, 0, stream>>>((float4*)cnt, N / 4);
  zero4_kernel<<<((N * 64) / 4 + B - 1) / B, B, 0, stream>>>((float4*)bufA,
                                                             (N * 64) / 4);
  degree_kernel<<<(E + B - 1) / B, B, 0, stream>>>(dst, cnt, E);
  scatter_kernel<64><<<(E * 16 + B - 1) / B, B, 0, stream>>>(x, src, dst, bufA, E);
  invcnt_kernel<<<(N + B - 1) / B, B, 0, stream>>>(cnt, N);
  {
    int waves = ((N + 15) / 16) * (128 / 16);
    sage_gemm_kernel<64, 128><<<(waves + 7) / 8, 256, 0, stream>>>(
        bufA, cnt, x, w1l, w1r, b1l, h1, N);
  }

  // ---- Layer 2 (transform-before-scatter) ----
  // yl = h1 @ w2l^T  (N x 64) -> scatter yl -> agg2 ; then fused epilogue.
  {
    int waves = ((N + 15) / 16) * (64 / 16);
    gemm_plain_kernel<128, 64, false><<<(waves + 7) / 8, 256, 0, stream>>>(
        h1, w2l, 128, nullptr, bufA, N);
  }
  zero4_kernel<<<((N * 64) / 4 + B - 1) / B, B, 0, stream>>>((float4*)agg2,
                                                             (N * 64) / 4);
  scatter_kernel<64><<<(E * 16 + B - 1) / B, B, 0, stream>>>(bufA, src, dst,
                                                             agg2, E);
  {
    int waves = ((N + 15) / 16) * (64 / 16);
    sage_gemm2_kernel<<<(waves + 7) / 8, 256, 0, stream>>>(h1, w2r, agg2, cnt,
                                                           b2l, h2, N);
  }

  // ---- Edge head: hoist MLP layer 1 to node level ----
  // u = h2 @ Wa^T + bc1,  v = h2 @ Wb^T   (wc1 = [Wa | Wb], row stride 128)
  {
    int waves = ((N + 15) / 16) * (64 / 16);
    gemm_plain_kernel<64, 64, true><<<(waves + 7) / 8, 256, 0, stream>>>(
        h2, wc1, 128, bc1, u, N);
    gemm_plain_kernel<64, 64, false><<<(waves + 7) / 8, 256, 0, stream>>>(
        h2, wc1 + 64, 128, nullptr, v, N);
  }
  // out[e] = relu(u[src]+v[dst]) @ wc2^T + bc2
  edge_head_kernel<<<(E + 7) / 8, 256, 0, stream>>>(u, v, src, dst, wc2, bc2,
                                                    out, E);
}